// FCOSDecoder_39350490366621
// MI455X (gfx1250) — compile-verified
//
#include <hip/hip_runtime.h>
#include <stdint.h>

// ---------------- problem constants ----------------
#define BATCH      16
#define NCLS       80
#define NLOC       21824           // 128^2 + 64^2 + 32^2 + 16^2 + 8^2
#define TOPN       1000
#define MIN_SCORE  0.05f
#define NMS_THR    0.6f
#define MAX_OBJ    100
#define CAND_CAP   2048            // pow2 >= TOPN (+ ties), bitonic size
#define MASK_W     32              // ceil(TOPN/32)

// workspace byte offsets (d_ws)
#define KEYS_BYTES   ((size_t)BATCH * NLOC * 8)           // 2,793,472
#define BOX_BYTES    ((size_t)BATCH * NLOC * 16)          // 5,586,944
#define CLSI_BYTES   ((size_t)BATCH * NLOC * 4)           // 1,396,736
#define MASK_OFF     (KEYS_BYTES + BOX_BYTES + CLSI_BYTES)

struct DecodeParams {
    const float* cls[5];
    const float* reg[5];
    const float* ctr[5];
    const float* pos[5];
    unsigned long long* keys;   // [B*NLOC] packed (score_bits<<32)|~loc, 0 = invalid
    float4*             boxes;  // [B*NLOC]
    int*                clsidx; // [B*NLOC]
};

// ---------------------------------------------------------------------------
// Kernel 1: decode. One wave32 per location. 32 lanes cover the 80-class row
// (coalesced 128B loads), wave shuffle argmax reduction, lane 0 decodes.
// ---------------------------------------------------------------------------
__global__ __launch_bounds__(256) void fcos_decode(DecodeParams p) {
    const int wavesPerBlock = blockDim.x >> 5;
    const int gw   = blockIdx.x * wavesPerBlock + (threadIdx.x >> 5);
    if (gw >= BATCH * NLOC) return;
    const int lane = threadIdx.x & 31;
    const int b    = gw / NLOC;
    const int loc  = gw - b * NLOC;

    int l, li, hw;
    if      (loc < 16384) { l = 0; li = loc;         hw = 16384; }
    else if (loc < 20480) { l = 1; li = loc - 16384; hw = 4096;  }
    else if (loc < 21504) { l = 2; li = loc - 20480; hw = 1024;  }
    else if (loc < 21760) { l = 3; li = loc - 21504; hw = 256;   }
    else                  { l = 4; li = loc - 21760; hw = 64;    }

    const size_t rbase = (size_t)b * hw + li;
    const float* crow  = p.cls[l] + rbase * NCLS;

    if (lane == 0) {                       // CDNA5 global_prefetch_b8
        __builtin_prefetch(p.reg[l] + rbase * 4, 0, 0);
        __builtin_prefetch(p.pos[l] + rbase * 2, 0, 0);
        __builtin_prefetch(p.ctr[l] + rbase,     0, 0);
    }

    // 80 = 32 + 32 + 16 lanes
    float v0 = crow[lane];
    float v1 = crow[lane + 32];
    float v2 = (lane < 16) ? crow[lane + 64] : -1.0f;

    float best = v0; int bidx = lane;
    if (v1 > best) { best = v1; bidx = lane + 32; }
    if (v2 > best) { best = v2; bidx = lane + 64; }

    // wave32 butterfly argmax (first-index tie-break, matches jnp.argmax)
    #pragma unroll
    for (int off = 16; off > 0; off >>= 1) {
        float ov = __shfl_xor(best, off, 32);
        int   oi = __shfl_xor(bidx, off, 32);
        if (ov > best || (ov == best && oi < bidx)) { best = ov; bidx = oi; }
    }

    if (lane == 0) {
        const float  t  = p.ctr[l][rbase];
        const float  sc = sqrtf(best * t);
        const float* pr = p.reg[l] + rbase * 4;
        const float* pp = p.pos[l] + rbase * 2;
        const float px = pp[0], py = pp[1];
        float4 bx;
        bx.x = truncf(px - pr[0]);
        bx.y = truncf(py - pr[1]);
        bx.z = truncf(px + pr[2]);
        bx.w = truncf(py + pr[3]);
        const size_t o = (size_t)b * NLOC + loc;
        p.boxes[o]  = bx;
        p.clsidx[o] = bidx;
        unsigned long long key = 0ull;
        if (sc > MIN_SCORE)
            key = ((unsigned long long)__float_as_uint(sc) << 32) |
                  (unsigned)(~(unsigned)loc);   // ~loc: smaller loc wins ties
        p.keys[o] = key;
    }
}

// ---------------------------------------------------------------------------
// Kernel 2: per-batch top-1000 (radix select + LDS bitonic sort), IoU mask
// matrix, serial greedy pass on one wave, scatter first 100 kept detections.
// One 1024-thread block per batch.
// ---------------------------------------------------------------------------
__global__ __launch_bounds__(1024) void fcos_topk_nms(
        const unsigned long long* __restrict__ keysWS,
        const float4* __restrict__ boxesWS,
        const int*    __restrict__ clsWS,
        unsigned*     __restrict__ maskWS,
        float*        __restrict__ out) {

    __shared__ unsigned long long cand[CAND_CAP];
    __shared__ float4   topBox[TOPN];
    __shared__ float    areas[TOPN];
    __shared__ unsigned hist[256];
    __shared__ unsigned sh_prefix, sh_maskhi, sh_remaining, sh_ncand;

    const int tid  = threadIdx.x;
    const int bdim = blockDim.x;
    const int b    = blockIdx.x;
    const unsigned long long* keys = keysWS + (size_t)b * NLOC;
    unsigned* maskG = maskWS + (size_t)b * TOPN * MASK_W;

    float* out_s = out + b * MAX_OBJ;
    float* out_c = out + BATCH * MAX_OBJ + b * MAX_OBJ;
    float* out_b = out + 2 * BATCH * MAX_OBJ + b * MAX_OBJ * 4;

    // default fill (overwritten by scatter later; barriers order same-address)
    for (int i = tid; i < MAX_OBJ; i += bdim) { out_s[i] = -1.0f; out_c[i] = -1.0f; }
    for (int i = tid; i < MAX_OBJ * 4; i += bdim) out_b[i] = 0.0f;

    if (tid == 0) { sh_prefix = 0u; sh_maskhi = 0u; sh_remaining = TOPN; sh_ncand = 0u; }
    __syncthreads();

    // ---- exact top-1000 threshold: 4-pass byte radix select over score bits
    for (int pass = 0; pass < 4; ++pass) {
        const int shift = 24 - pass * 8;
        for (int i = tid; i < 256; i += bdim) hist[i] = 0u;
        __syncthreads();
        const unsigned pmask = sh_maskhi, pval = sh_prefix;
        for (int i = tid; i < NLOC; i += bdim) {
            const unsigned s = (unsigned)(keys[i] >> 32);
            if ((s & pmask) == pval)
                atomicAdd(&hist[(s >> shift) & 0xFF], 1u);
        }
        __syncthreads();
        if (tid == 0) {
            const unsigned rem = sh_remaining;
            unsigned cum = 0; int t = 255;
            for (; t > 0; --t) {
                const unsigned h = hist[t];
                if (cum + h >= rem) break;
                cum += h;
            }
            sh_prefix    = pval  | ((unsigned)t << shift);
            sh_maskhi    = pmask | (0xFFu << shift);
            sh_remaining = rem - cum;
        }
        __syncthreads();
    }
    const unsigned T = sh_prefix;

    // ---- compact candidates with score_bits >= T into LDS
    for (int i = tid; i < NLOC; i += bdim) {
        const unsigned long long k = keys[i];
        if ((unsigned)(k >> 32) >= T) {
            const unsigned pos = atomicAdd(&sh_ncand, 1u);
            if (pos < CAND_CAP) cand[pos] = k;
        }
    }
    __syncthreads();
    const unsigned nc = (sh_ncand < CAND_CAP) ? sh_ncand : CAND_CAP;
    for (int i = tid; i < CAND_CAP; i += bdim)
        if ((unsigned)i >= nc) cand[i] = 0ull;
    __syncthreads();

    // ---- bitonic sort descending, 2048 x u64 in LDS
    for (unsigned k = 2; k <= CAND_CAP; k <<= 1) {
        for (unsigned j = k >> 1; j > 0; j >>= 1) {
            for (unsigned i = tid; i < CAND_CAP; i += bdim) {
                const unsigned ixj = i ^ j;
                if (ixj > i) {
                    const bool descend = ((i & k) == 0);
                    const unsigned long long a = cand[i], c = cand[ixj];
                    if (descend ? (a < c) : (a > c)) { cand[i] = c; cand[ixj] = a; }
                }
            }
            __syncthreads();
        }
    }

    // ---- gather top-1000 boxes into LDS via CDNA5 async global->LDS b128
    for (int j = tid; j < TOPN; j += bdim) {
        const unsigned long long k = cand[j];
        const float sc = __uint_as_float((unsigned)(k >> 32));
        if (sc > MIN_SCORE) {
            const unsigned loc = ~(unsigned)k;
            const unsigned long long gaddr =
                (unsigned long long)(const void*)(boxesWS + (size_t)b * NLOC + loc);
            const unsigned ldsoff = (unsigned)(size_t)(const void*)(&topBox[j]);
            asm volatile("global_load_async_to_lds_b128 %0, %1, off"
                         :: "v"(ldsoff), "v"(gaddr) : "memory");
        } else {
            topBox[j] = make_float4(0.f, 0.f, 0.f, 0.f);
        }
    }
    asm volatile("s_wait_asynccnt 0" ::: "memory");
    __syncthreads();

    for (int j = tid; j < TOPN; j += bdim) {
        const float4 bb = topBox[j];
        areas[j] = fmaxf((bb.z - bb.x) * (bb.w - bb.y), 1e-4f);
    }
    __syncthreads();

    // ---- parallel 1000x1000 IoU suppression bit-matrix (upper triangle)
    for (int item = tid; item < TOPN * MASK_W; item += bdim) {
        const int i = item >> 5;
        const int w = item & (MASK_W - 1);
        const float4 bi = topBox[i];
        const float  ai = areas[i];
        unsigned bits = 0u;
        const int jbase = w << 5;
        #pragma unroll 4
        for (int t = 0; t < 32; ++t) {
            const int j = jbase + t;
            if (j < TOPN && j > i) {
                const float4 bj = topBox[j];
                const float ow = fmaxf(fminf(bi.z, bj.z) - fmaxf(bi.x, bj.x), 0.f);
                const float oh = fmaxf(fminf(bi.w, bj.w) - fmaxf(bi.y, bj.y), 0.f);
                const float inter = ow * oh;
                const float uni   = fmaxf(ai + areas[j] - inter, 1e-4f);
                if (inter >= NMS_THR * uni) bits |= (1u << t);
            }
        }
        maskG[item] = bits;        // item == i*32 + w
    }
    __syncthreads();

    // ---- serial greedy pass + output scatter on wave 0 only
    if (tid < 32) {
        const int lane = tid;
        unsigned rem = 0u;         // lane owns suppressed bits [32*lane, 32*lane+32)
        for (int i = 0; i < TOPN; ++i) {
            const unsigned remw = __shfl(rem, i >> 5, 32);
            const bool sup = (remw >> (i & 31)) & 1u;
            const float sc = __uint_as_float((unsigned)(cand[i] >> 32));
            const unsigned mrow = maskG[(size_t)i * MASK_W + lane];
            if ((sc > MIN_SCORE) && !sup) rem |= mrow;
        }

        // keep mask per lane word
        unsigned validw = 0u;
        for (int t = 0; t < 32; ++t) {
            const int j = (lane << 5) + t;
            if (j < TOPN) {
                const float sc = __uint_as_float((unsigned)(cand[j] >> 32));
                if (sc > MIN_SCORE) validw |= (1u << t);
            }
        }
        const unsigned keepw = validw & ~rem;

        // exclusive prefix of popcounts across lanes
        const unsigned cnt = __popc(keepw);
        unsigned incl = cnt;
        #pragma unroll
        for (int d = 1; d < 32; d <<= 1) {
            const unsigned n = __shfl_up(incl, d, 32);
            if (lane >= d) incl += n;
        }
        const unsigned base = incl - cnt;

        for (int t = 0; t < 32; ++t) {
            if (keepw & (1u << t)) {
                const unsigned r = base + __popc(keepw & ((1u << t) - 1u));
                if (r < MAX_OBJ) {
                    const int j = (lane << 5) + t;
                    const unsigned long long k = cand[j];
                    const unsigned loc = ~(unsigned)k;
                    out_s[r] = __uint_as_float((unsigned)(k >> 32));
                    out_c[r] = (float)clsWS[(size_t)b * NLOC + loc];
                    const float4 bb = topBox[j];
                    out_b[r * 4 + 0] = bb.x;
                    out_b[r * 4 + 1] = bb.y;
                    out_b[r * 4 + 2] = bb.z;
                    out_b[r * 4 + 3] = bb.w;
                }
            }
        }
    }
}

// ---------------------------------------------------------------------------
extern "C" void kernel_launch(void* const* d_in, const int* in_sizes, int n_in,
                              void* d_out, int out_size, void* d_ws, size_t ws_size,
                              hipStream_t stream) {
    (void)in_sizes; (void)n_in; (void)out_size; (void)ws_size;

    DecodeParams P;
    for (int l = 0; l < 5; ++l) {
        P.cls[l] = (const float*)d_in[4 * l + 0];
        P.reg[l] = (const float*)d_in[4 * l + 1];
        P.ctr[l] = (const float*)d_in[4 * l + 2];
        P.pos[l] = (const float*)d_in[4 * l + 3];
    }
    char* ws = (char*)d_ws;
    unsigned long long* keys  = (unsigned long long*)ws;
    float4*             boxes = (float4*)(ws + KEYS_BYTES);
    int*                clsix = (int*)(ws + KEYS_BYTES + BOX_BYTES);
    unsigned*           mask  = (unsigned*)(ws + MASK_OFF);
    P.keys = keys; P.boxes = boxes; P.clsidx = clsix;

    const int totalWaves = BATCH * NLOC;           // one wave32 per location
    const int wpb = 256 / 32;
    const int grid1 = (totalWaves + wpb - 1) / wpb;
    fcos_decode<<<grid1, 256, 0, stream>>>(P);

    fcos_topk_nms<<<BATCH, 1024, 0, stream>>>(keys, boxes, clsix, mask, (float*)d_out);
}